// SLDS_43903155699931
// MI455X (gfx1250) — compile-verified
//
#include <hip/hip_runtime.h>
#include <hip/hip_bf16.h>

// SLDS GPB1 filter for MI455X (gfx1250, wave32).
// All dense GEMMs use V_WMMA_F32_16X16X4_F32 (fp32 WMMA, no precision loss).
// K=16 components -> 16 workgroups; D=64 state; O=32 obs; sequential over T.

typedef float v2f __attribute__((ext_vector_type(2)));
typedef float v8f __attribute__((ext_vector_type(8)));

#define SLDS_LOG2PI 1.8378770664093453f

// ---- WMMA 16x16 fp32 tile GEMM: acc += A(16xK) * B(Kx16) ----------------
// A fragment layout (16x4 fp32, ISA 7.12.2): lanes 0-15 hold M=0..15,
//   vgpr0 = K=kbase+0, vgpr1 = K=kbase+1, with kbase = (lane/16)*2.
// B fragment mirrors with N across lanes. C/D: vgpr v, lane half h ->
//   row v + 8h, col lane%16.
template <bool TA, bool TB>
__device__ __forceinline__ v8f wmma_tile(v8f acc,
                                         const float* __restrict__ A, int lda,
                                         const float* __restrict__ B, int ldb,
                                         int K) {
  const int lane = threadIdx.x & 31;
  const int l16  = lane & 15;
  const int kh   = (lane >> 4) << 1;  // 0 or 2
#pragma unroll 4
  for (int k0 = 0; k0 < K; k0 += 4) {
    const int ka = k0 + kh;
    v2f a, b;
    if (TA) { a.x = A[(ka + 0) * lda + l16]; a.y = A[(ka + 1) * lda + l16]; }
    else    { a.x = A[l16 * lda + ka + 0];   a.y = A[l16 * lda + ka + 1]; }
    if (TB) { b.x = B[l16 * ldb + ka + 0];   b.y = B[l16 * ldb + ka + 1]; }
    else    { b.x = B[(ka + 0) * ldb + l16]; b.y = B[(ka + 1) * ldb + l16]; }
    acc = __builtin_amdgcn_wmma_f32_16x16x4_f32(false, a, false, b,
                                                (short)0, acc, false, false);
  }
  return acc;
}

__device__ __forceinline__ void store_tile(float* dst, int ld, v8f acc) {
  const int lane  = threadIdx.x & 31;
  const int l16   = lane & 15;
  const int rbase = (lane >> 4) * 8;
#pragma unroll
  for (int v = 0; v < 8; ++v) dst[(rbase + v) * ld + l16] = acc[v];
}

// ---- prep: logA = log_softmax(rows), Q=exp, R=exp, init state -----------
__global__ __launch_bounds__(256) void slds_prep(
    const float* __restrict__ tl, const float* __restrict__ ltn,
    const float* __restrict__ lon, float* __restrict__ logA,
    float* __restrict__ Qd, float* __restrict__ Rd, float* __restrict__ w0,
    float* __restrict__ m0, float* __restrict__ P0) {
  const int tid = threadIdx.x;
  if (tid < 16) {
    float mx = -3.4e38f;
    for (int c = 0; c < 16; ++c) mx = fmaxf(mx, tl[tid * 16 + c]);
    float s = 0.f;
    for (int c = 0; c < 16; ++c) s += expf(tl[tid * 16 + c] - mx);
    const float ls = mx + logf(s);
    for (int c = 0; c < 16; ++c) logA[tid * 16 + c] = tl[tid * 16 + c] - ls;
  }
  if (tid < 64) Qd[tid] = expf(ltn[tid]);
  if (tid < 32) Rd[tid] = expf(lon[tid]);
  for (int idx = tid; idx < 16 * 64; idx += 256) m0[idx] = 0.f;
  for (int idx = tid; idx < 16 * 64 * 64; idx += 256) {
    const int r = (idx >> 6) & 63, c = idx & 63;
    P0[idx] = (r == c) ? 1.f : 0.f;
  }
  __syncthreads();
  if (tid < 16) w0[tid] = logA[tid];  // row 0 of logA
}

// ---- predict + moment-match collapse (one block per target component j) --
__global__ __launch_bounds__(256) void slds_predict(
    const float* __restrict__ w_in, const float* __restrict__ m_in,
    const float* __restrict__ P_in, const float* __restrict__ logA,
    const float* __restrict__ Mg, const float* __restrict__ Qg,
    float* __restrict__ w_out, float* __restrict__ m_out,
    float* __restrict__ P_out) {
  const int j = blockIdx.x;
  const int tid = threadIdx.x;
  const int wave = tid >> 5;

  __shared__ float sM[64 * 64];    // M_j
  __shared__ float sPbar[64 * 64]; // sum_i r_i P_i
  __shared__ float sT1[64 * 64];   // M_j^T Pbar
  __shared__ float sMu[16 * 64];   // mu, then sqrt(r)*mu
  __shared__ float sm[16 * 64];    // prior means
  __shared__ float sR[16];
  __shared__ float sMnew[64];
  __shared__ float sQ[64];
  __shared__ float sWnew;

  if (tid == 0) {  // responsibilities + new log weight
    float l[16], mx = -3.4e38f;
    for (int i = 0; i < 16; ++i) {
      l[i] = w_in[i] + logA[i * 16 + j];
      mx = fmaxf(mx, l[i]);
    }
    float s = 0.f;
    for (int i = 0; i < 16; ++i) s += expf(l[i] - mx);
    const float wn = mx + logf(s);
    sWnew = wn;
    for (int i = 0; i < 16; ++i) sR[i] = expf(l[i] - wn);
  }
  for (int idx = tid; idx < 64 * 64; idx += 256) sM[idx] = Mg[j * 4096 + idx];
  for (int idx = tid; idx < 16 * 64; idx += 256) sm[idx] = m_in[idx];
  if (tid < 64) sQ[tid] = Qg[tid];
  __syncthreads();

  // Pbar = sum_i r_i P_i  (collapse BEFORE the sandwich: M_j shared over i)
  for (int idx = tid; idx < 4096; idx += 256) {
    float acc = 0.f;
#pragma unroll
    for (int i = 0; i < 16; ++i) acc += sR[i] * P_in[i * 4096 + idx];
    sPbar[idx] = acc;
  }
  // mu = m @ M_j  (16x64), waves 0..3
  if (wave < 4) {
    const int n0 = wave * 16;
    v8f acc = {};
    acc = wmma_tile<false, false>(acc, sm, 64, sM + n0, 64, 64);
    store_tile(sMu + n0, 64, acc);
  }
  __syncthreads();

  if (tid < 64) {  // m_new = r^T mu
    float mn = 0.f;
#pragma unroll
    for (int i = 0; i < 16; ++i) mn += sR[i] * sMu[i * 64 + tid];
    sMnew[tid] = mn;
    m_out[j * 64 + tid] = mn;
  }
  __syncthreads();

  // U = sqrt(r_i) * mu_i (in place);  T1 = M_j^T Pbar
  for (int idx = tid; idx < 1024; idx += 256) sMu[idx] *= sqrtf(sR[idx >> 6]);
  {
    const int t = wave;  // tiles t and t+8
#pragma unroll
    for (int s = 0; s < 2; ++s) {
      const int tt = t + s * 8;
      const int m0 = (tt >> 2) * 16, n0 = (tt & 3) * 16;
      v8f acc = {};
      acc = wmma_tile<true, false>(acc, sM + m0, 64, sPbar + n0, 64, 64);
      store_tile(sT1 + m0 * 64 + n0, 64, acc);
    }
  }
  __syncthreads();

  // P_new = T1 @ M_j + U^T U - m m^T + diag(Q)
  {
    const int lane = threadIdx.x & 31;
    const int l16 = lane & 15;
    const int rbase = (lane >> 4) * 8;
#pragma unroll
    for (int s = 0; s < 2; ++s) {
      const int tt = wave + s * 8;
      const int m0 = (tt >> 2) * 16, n0 = (tt & 3) * 16;
      v8f acc = {};
      acc = wmma_tile<true, false>(acc, sMu + m0, 64, sMu + n0, 64, 16);
      acc = wmma_tile<false, false>(acc, sT1 + m0 * 64, 64, sM + n0, 64, 64);
#pragma unroll
      for (int v = 0; v < 8; ++v) {
        const int row = m0 + rbase + v, col = n0 + l16;
        float val = acc[v] - sMnew[row] * sMnew[col];
        if (row == col) val += sQ[col];
        P_out[j * 4096 + row * 64 + col] = val;
      }
    }
  }
  if (tid == 0) w_out[j] = sWnew;
}

// ---- measurement update (in place), one block per component j -----------
__global__ __launch_bounds__(256) void slds_observe(
    float* __restrict__ w, float* __restrict__ m, float* __restrict__ P,
    const float* __restrict__ Cg, const float* __restrict__ Rg,
    const float* __restrict__ y) {
  const int j = blockIdx.x;
  const int tid = threadIdx.x;
  const int wave = tid >> 5;
  const int lane = tid & 31;

  __shared__ float sP[64 * 64];   // P_j
  __shared__ float sC[64 * 32];   // C
  __shared__ float sPC[64 * 32];  // P C
  __shared__ float sL[32 * 33];   // S then its Cholesky factor (ld=33)
  __shared__ float sX[32 * 64];   // S^{-1} PC^T  (Kg transposed)
  __shared__ float sMv[64], sInnov[32], sZ[32], sLog[32];

  for (int idx = tid; idx < 4096; idx += 256) sP[idx] = P[j * 4096 + idx];
  for (int idx = tid; idx < 2048; idx += 256) sC[idx] = Cg[idx];
  if (tid < 64) sMv[tid] = m[j * 64 + tid];
  __syncthreads();

  // PC = P @ C : 64x32, 8 tiles
  {
    const int m0 = (wave >> 1) * 16, n0 = (wave & 1) * 16;
    v8f acc = {};
    acc = wmma_tile<false, false>(acc, sP + m0 * 64, 64, sC + n0, 32, 64);
    store_tile(sPC + m0 * 32 + n0, 32, acc);
  }
  __syncthreads();

  // S = C^T PC (waves 0..3); innov = y - m C (wave 4)
  if (wave < 4) {
    const int m0 = (wave >> 1) * 16, n0 = (wave & 1) * 16;
    v8f acc = {};
    acc = wmma_tile<true, false>(acc, sC + m0, 32, sPC + n0, 32, 64);
    store_tile(sL + m0 * 33 + n0, 33, acc);
  } else if (wave == 4) {
    float acc = 0.f;
    for (int d = 0; d < 64; ++d) acc += sMv[d] * sC[d * 32 + lane];
    sInnov[lane] = y[lane] - acc;
  }
  __syncthreads();
  if (tid < 32) sL[tid * 33 + tid] += Rg[tid];
  __syncthreads();

  // Cholesky (lower) + solves, single wave (lockstep => in-order LDS)
  if (tid < 32) {
    const int r = lane;
    for (int k = 0; k < 32; ++k) {
      const float dk = sqrtf(sL[k * 33 + k]);
      float lrk = 0.f;
      if (r == k) sL[k * 33 + k] = dk;
      if (r > k) { lrk = sL[r * 33 + k] / dk; sL[r * 33 + k] = lrk; }
      __builtin_amdgcn_wave_barrier();
      for (int q = k + 1; q < 32; ++q)
        if (r >= q) sL[r * 33 + q] -= lrk * sL[q * 33 + k];
      __builtin_amdgcn_wave_barrier();
    }
    sLog[r] = logf(sL[r * 33 + r]);
    // z = S^{-1} innov : forward then backward substitution
    sZ[r] = sInnov[r];
    __builtin_amdgcn_wave_barrier();
    for (int k = 0; k < 32; ++k) {
      if (r == k) sZ[k] /= sL[k * 33 + k];
      __builtin_amdgcn_wave_barrier();
      const float vk = sZ[k];
      if (r > k) sZ[r] -= sL[r * 33 + k] * vk;
      __builtin_amdgcn_wave_barrier();
    }
    for (int k = 31; k >= 0; --k) {
      if (r == k) sZ[k] /= sL[k * 33 + k];
      __builtin_amdgcn_wave_barrier();
      const float vk = sZ[k];
      if (r < k) sZ[r] -= sL[k * 33 + r] * vk;
      __builtin_amdgcn_wave_barrier();
    }
  }
  __syncthreads();

  if (tid == 0) {  // log-likelihood -> weight
    float sl = 0.f, quad = 0.f;
    for (int o = 0; o < 32; ++o) { sl += sLog[o]; quad += sZ[o] * sInnov[o]; }
    w[j] += -0.5f * (32.f * SLDS_LOG2PI + 2.f * sl + quad);
  }

  // X = S^{-1} PC^T (32x64): one thread per RHS column c
  if (tid < 64) {
    const int c = tid;
    for (int k = 0; k < 32; ++k) {
      float s = sPC[c * 32 + k];
      for (int q = 0; q < k; ++q) s -= sL[k * 33 + q] * sX[q * 64 + c];
      sX[k * 64 + c] = s / sL[k * 33 + k];
    }
    for (int k = 31; k >= 0; --k) {
      float s = sX[k * 64 + c];
      for (int q = k + 1; q < 32; ++q) s -= sL[q * 33 + k] * sX[q * 64 + c];
      sX[k * 64 + c] = s / sL[k * 33 + k];
    }
  }
  __syncthreads();

  if (tid < 64) {  // m += Kg innov  (Kg[d][o] = X[o][d])
    float acc = sMv[tid];
    for (int o = 0; o < 32; ++o) acc += sX[o * 64 + tid] * sInnov[o];
    m[j * 64 + tid] = acc;
  }
  // P -= Kg @ PC^T : 16 tiles, WMMA K=32
  {
    const int l16 = lane & 15;
    const int rbase = (lane >> 4) * 8;
#pragma unroll
    for (int s = 0; s < 2; ++s) {
      const int tt = wave + s * 8;
      const int m0 = (tt >> 2) * 16, n0 = (tt & 3) * 16;
      v8f acc = {};
      acc = wmma_tile<true, true>(acc, sX + m0, 64, sPC + n0 * 32, 32, 32);
#pragma unroll
      for (int v = 0; v < 8; ++v) {
        const int row = m0 + rbase + v, col = n0 + l16;
        P[j * 4096 + row * 64 + col] = sP[row * 64 + col] - acc[v];
      }
    }
  }
}

__global__ void slds_final(const float* __restrict__ w, float* __restrict__ out) {
  if (threadIdx.x == 0) {
    float mx = -3.4e38f;
    for (int i = 0; i < 16; ++i) mx = fmaxf(mx, w[i]);
    float s = 0.f;
    for (int i = 0; i < 16; ++i) s += expf(w[i] - mx);
    out[0] = mx + logf(s);
  }
}

extern "C" void kernel_launch(void* const* d_in, const int* in_sizes, int n_in,
                              void* d_out, int out_size, void* d_ws,
                              size_t ws_size, hipStream_t stream) {
  const float* data = (const float*)d_in[0];  // (T,O)
  const float* tl   = (const float*)d_in[1];  // (K,K)
  const float* Mg   = (const float*)d_in[2];  // (K,D,D)
  const float* ltn  = (const float*)d_in[3];  // (D,)
  const float* Cg   = (const float*)d_in[4];  // (D,O)
  const float* lon  = (const float*)d_in[5];  // (O,)
  const int K = 16, D = 64, O = 32;
  const int T = in_sizes[0] / O;

  float* ws   = (float*)d_ws;
  float* logA = ws;                 // K*K
  float* Qd   = logA + K * K;       // D
  float* Rd   = Qd + D;             // O
  float* bufA = Rd + O;
  const int statesz = K + K * D + K * D * D;
  float* bufB = bufA + statesz;
  float *wA = bufA, *mA = bufA + K, *PA = mA + K * D;
  float *wB = bufB, *mB = bufB + K, *PB = mB + K * D;

  slds_prep<<<1, 256, 0, stream>>>(tl, ltn, lon, logA, Qd, Rd, wA, mA, PA);
  slds_observe<<<K, 256, 0, stream>>>(wA, mA, PA, Cg, Rd, data);

  float *wc = wA, *mc = mA, *Pc = PA;
  float *wo = wB, *mo = mB, *Po = PB;
  for (int t = 1; t < T; ++t) {
    slds_predict<<<K, 256, 0, stream>>>(wc, mc, Pc, logA, Mg, Qd, wo, mo, Po);
    slds_observe<<<K, 256, 0, stream>>>(wo, mo, Po, Cg, Rd, data + t * O);
    float* tmp;
    tmp = wc; wc = wo; wo = tmp;
    tmp = mc; mc = mo; mo = tmp;
    tmp = Pc; Pc = Po; Po = tmp;
  }
  slds_final<<<1, 32, 0, stream>>>(wc, (float*)d_out);
}